// FocalLoss_39883066311339
// MI455X (gfx1250) — compile-verified
//
#include <hip/hip_runtime.h>
#include <stdint.h>

// ---------------------------------------------------------------------------
// RetinaNet focal loss on MI455X (gfx1250).
// Memory-bound streaming kernel: 307MB of classifications read once.
// Uses CDNA5 async global->LDS copies (ASYNCcnt) to stage the class tile
// while the IoU / assignment / regression math runs, then s_wait_asynccnt.
// ---------------------------------------------------------------------------

#define TILE_A          256                 // anchors per block (1 thread = 1 anchor)
#define NCLASS          80
#define ROWW            84                  // padded LDS row (words): 336B, 16B aligned
#define CHUNKS_PER_ROW  (NCLASS / 4)        // 20 x b128 per anchor row
#define MAXM            32

__global__ void fl_zero_ws(float* ws, int n) {
    int i = blockIdx.x * blockDim.x + threadIdx.x;
    if (i < n) ws[i] = 0.0f;
}

__global__ __launch_bounds__(TILE_A) void fl_main(
    const float* __restrict__ cls,   // [B, A, 80]
    const float* __restrict__ reg,   // [B, A, 4]
    const float* __restrict__ anc,   // [1, A, 4]
    const float* __restrict__ ann,   // [B, M, 5]
    float* __restrict__ ws,          // [B, 3]: cls_sum, reg_sum, num_pos
    int A, int M, int blocksPerImage)
{
    __shared__ float tile[TILE_A * ROWW];   // 86016 B
    __shared__ float s_ann[7 * MAXM];       // x1,y1,x2,y2,cls,area,valid

    const int b   = blockIdx.x / blocksPerImage;
    const int a0  = (blockIdx.x % blocksPerImage) * TILE_A;
    const int tid = (int)threadIdx.x;
    const int a   = a0 + tid;

    // ---- kick async global->LDS staging of this block's classification tile ----
    {
        const float* gbase = cls + ((size_t)b * (size_t)A + (size_t)a0) * NCLASS;
        uint32_t lds0 = (uint32_t)(uintptr_t)(&tile[0]);   // low 32 bits = LDS offset
        int rows = A - a0; if (rows > TILE_A) rows = TILE_A;
        int nChunks = rows * CHUNKS_PER_ROW;
#pragma unroll
        for (int it = 0; it < CHUNKS_PER_ROW; ++it) {
            int id = it * TILE_A + tid;
            if (id < nChunks) {
                int row = id / CHUNKS_PER_ROW;
                int k   = id % CHUNKS_PER_ROW;
                uint32_t ldsb = lds0 + (uint32_t)row * (uint32_t)(ROWW * 4) + (uint32_t)k * 16u;
                uint32_t goff = (uint32_t)id * 16u;        // contiguous global layout
                asm volatile("global_load_async_to_lds_b128 %0, %1, %2"
                             :: "v"(ldsb), "v"(goff), "s"(gbase) : "memory");
            }
        }
    }

    // ---- stage annotations (tiny) while async copies fly ----
    if (tid < M && tid < MAXM) {
        const float* g = ann + ((size_t)b * (size_t)M + (size_t)tid) * 5;
        float x1 = g[0], y1 = g[1], x2 = g[2], y2 = g[3], c4 = g[4];
        s_ann[0 * MAXM + tid] = x1;
        s_ann[1 * MAXM + tid] = y1;
        s_ann[2 * MAXM + tid] = x2;
        s_ann[3 * MAXM + tid] = y2;
        s_ann[4 * MAXM + tid] = c4;
        s_ann[5 * MAXM + tid] = (x2 - x1) * (y2 - y1);
        s_ann[6 * MAXM + tid] = (c4 != -1.0f) ? 1.0f : 0.0f;
    }
    __syncthreads();   // annotations visible (does not wait on async copies)

    // ---- IoU max/argmax, positives, regression loss (overlaps async copy) ----
    float csum = 0.0f, rsum = 0.0f, posf = 0.0f;
    const bool active = (a < A);
    bool pos = false, ignore = false;
    int  arg = 0;
    float best = -2.0f;
    float4 av = make_float4(0.f, 0.f, 0.f, 0.f);

    if (active) {
        av = ((const float4*)anc)[a];
        float aw = av.z - av.x, ah = av.w - av.y;
        float acx = av.x + 0.5f * aw, acy = av.y + 0.5f * ah;
        float areaA = aw * ah;

        for (int m = 0; m < M; ++m) {
            float bx1 = s_ann[0 * MAXM + m], by1 = s_ann[1 * MAXM + m];
            float bx2 = s_ann[2 * MAXM + m], by2 = s_ann[3 * MAXM + m];
            float ba  = s_ann[5 * MAXM + m], bv  = s_ann[6 * MAXM + m];
            float iw = fmaxf(fminf(av.z, bx2) - fmaxf(av.x, bx1), 0.0f);
            float ih = fmaxf(fminf(av.w, by2) - fmaxf(av.y, by1), 0.0f);
            float inter = iw * ih;
            float ua = fmaxf(areaA + ba - inter, 1e-8f);
            float iou = inter / ua;
            iou = (bv > 0.0f) ? iou : -1.0f;
            if (iou > best) { best = iou; arg = m; }   // first-max, like jnp.argmax
        }
        pos    = (best >= 0.5f);
        ignore = (!pos) && (best >= 0.4f);

        if (pos) {
            posf = 1.0f;
            float gx1 = s_ann[0 * MAXM + arg], gy1 = s_ann[1 * MAXM + arg];
            float gx2 = s_ann[2 * MAXM + arg], gy2 = s_ann[3 * MAXM + arg];
            float gw = gx2 - gx1, gh = gy2 - gy1;
            float gcx = gx1 + 0.5f * gw, gcy = gy1 + 0.5f * gh;  // before clamp
            gw = fmaxf(gw, 1.0f); gh = fmaxf(gh, 1.0f);
            float t0 = ((gcx - acx) / aw) * 10.0f;   // /0.1
            float t1 = ((gcy - acy) / ah) * 10.0f;
            float t2 = __logf(gw / aw) * 5.0f;       // /0.2
            float t3 = __logf(gh / ah) * 5.0f;
            float4 rp = ((const float4*)reg)[(size_t)b * (size_t)A + (size_t)a];
            float d;
            d = fabsf(t0 - rp.x); rsum += (d <= (1.0f/9.0f)) ? 4.5f*d*d : d - (0.5f/9.0f);
            d = fabsf(t1 - rp.y); rsum += (d <= (1.0f/9.0f)) ? 4.5f*d*d : d - (0.5f/9.0f);
            d = fabsf(t2 - rp.z); rsum += (d <= (1.0f/9.0f)) ? 4.5f*d*d : d - (0.5f/9.0f);
            d = fabsf(t3 - rp.w); rsum += (d <= (1.0f/9.0f)) ? 4.5f*d*d : d - (0.5f/9.0f);
        }
    }

    // ---- classification tile now required: drain ASYNCcnt, then barrier ----
    asm volatile("s_wait_asynccnt 0x0" ::: "memory");
    __syncthreads();

    if (active && !ignore) {
        int ac = pos ? (int)s_ann[4 * MAXM + arg] : -1;
        const float4* row = (const float4*)&tile[tid * ROWW];
#pragma unroll 4
        for (int c4 = 0; c4 < NCLASS / 4; ++c4) {
            float4 pv = row[c4];
            float pp[4] = { pv.x, pv.y, pv.z, pv.w };
            int cbase = c4 * 4;
#pragma unroll
            for (int j = 0; j < 4; ++j) {
                float p = fminf(fmaxf(pp[j], 1e-4f), 1.0f - 1e-4f);
                if (cbase + j == ac) {
                    float q = 1.0f - p;
                    csum += 0.25f * q * q * (-__logf(p));       // target == 1
                } else {
                    csum += 0.75f * p * p * (-__logf(1.0f - p)); // target == 0
                }
            }
        }
    }

    // ---- wave32 reduction, then per-image atomics ----
    float vals[3] = { csum, rsum, posf };
#pragma unroll
    for (int i = 0; i < 3; ++i) {
        float x = vals[i];
        for (int off = 16; off > 0; off >>= 1) x += __shfl_xor(x, off, 32);
        if ((threadIdx.x & 31u) == 0u) atomicAdd(&ws[b * 3 + i], x);
    }
}

__global__ void fl_finalize(const float* __restrict__ ws, float* __restrict__ out, int B) {
    if (blockIdx.x == 0 && threadIdx.x == 0) {
        float cm = 0.0f, rm = 0.0f;
        for (int b = 0; b < B; ++b) {
            float np = ws[b * 3 + 2];
            float cl = ws[b * 3 + 0] / fmaxf(np, 1.0f);
            float rl = (np > 0.0f) ? (ws[b * 3 + 1] / fmaxf(4.0f * np, 1.0f)) : 0.0f;
            cm += cl; rm += rl;
        }
        out[0] = cm / (float)B;
        out[1] = rm / (float)B;
    }
}

extern "C" void kernel_launch(void* const* d_in, const int* in_sizes, int n_in,
                              void* d_out, int out_size, void* d_ws, size_t ws_size,
                              hipStream_t stream) {
    const float* cls = (const float*)d_in[0];   // [B, A, 80]
    const float* reg = (const float*)d_in[1];   // [B, A, 4]
    const float* anc = (const float*)d_in[2];   // [1, A, 4]
    const float* ann = (const float*)d_in[3];   // [B, M, 5]
    float* out = (float*)d_out;
    float* ws  = (float*)d_ws;

    int A = in_sizes[2] / 4;
    int B = in_sizes[1] / (4 * A);
    int M = in_sizes[3] / (5 * B);
    if (M > MAXM) M = MAXM;

    fl_zero_ws<<<1, 64, 0, stream>>>(ws, 3 * B);

    int blocksPerImage = (A + TILE_A - 1) / TILE_A;
    fl_main<<<B * blocksPerImage, TILE_A, 0, stream>>>(cls, reg, anc, ann, ws,
                                                       A, M, blocksPerImage);

    fl_finalize<<<1, 32, 0, stream>>>(ws, out, B);
}